// GCNModel_386547056688
// MI455X (gfx1250) — compile-verified
//
#include <hip/hip_runtime.h>

typedef float v2f __attribute__((ext_vector_type(2)));
typedef float v8f __attribute__((ext_vector_type(8)));

#define IN_F  128
#define HID_F 128
#define OUT_F 64
#define KDIM  128   // all three layers have K = 128

// ---------------------------------------------------------------------------
// Utility kernels
// ---------------------------------------------------------------------------
__global__ void zero_kernel(float* __restrict__ p, long n) {
    long i = (long)blockIdx.x * blockDim.x + threadIdx.x;
    if (i < n) p[i] = 0.0f;
}

__global__ void degree_kernel(const int* __restrict__ src, const int* __restrict__ dst,
                              float* deg_src, float* deg_dst, int ne) {
    int e = blockIdx.x * blockDim.x + threadIdx.x;
    if (e < ne) {
        atomicAdd(&deg_src[src[e]], 1.0f);
        atomicAdd(&deg_dst[dst[e]], 1.0f);
    }
}

__global__ void rsqrt_kernel(float* __restrict__ p, int n) {
    int i = blockIdx.x * blockDim.x + threadIdx.x;
    if (i < n) {
        float d = p[i];
        p[i] = rsqrtf(d < 1.0f ? 1.0f : d);
    }
}

// y[i][f] = (relu?) (y[i][f] * dinv[i] + bias[f])
__global__ void epilogue_kernel(float* __restrict__ Y, const float* __restrict__ dinv,
                                const float* __restrict__ bias, int F, long total,
                                int do_relu) {
    long t = (long)blockIdx.x * blockDim.x + threadIdx.x;
    if (t >= total) return;
    int i = (int)(t / F);
    int f = (int)(t - (long)i * F);
    float v = Y[t] * dinv[i] + bias[f];
    if (do_relu) v = v > 0.0f ? v : 0.0f;
    Y[t] = v;
}

// ---------------------------------------------------------------------------
// H[16-row block] = (X * dinv) @ W   using V_WMMA_F32_16X16X4_F32
//   X: [N, 128] row major, W: [128, FOUT] row major, H: [N, FOUT]
//   grid.x = N/16 (N divisible by 16), block = (FOUT/16) waves * 32 threads
//   Wave w computes the 16x16 C tile at columns [16w, 16w+16).
//   Per ISA 7.12.2 (f32 layouts):
//     A 16x4 : lane L -> m = L%16 ; VGPR j holds A[m][k0 + j + 2*(L/16)]
//     B 4x16 : lane L -> n = L%16 ; VGPR j holds B[k0 + j + 2*(L/16)][n]
//     C 16x16: lane L -> n = L%16 ; VGPR r holds C[r + 8*(L/16)][n]
//   No divergence anywhere: EXEC is all ones around every WMMA.
// ---------------------------------------------------------------------------
template <int FOUT>
__launch_bounds__(FOUT * 2)
__global__ void gemm_rows16_wmma(const float* __restrict__ X,
                                 const float* __restrict__ dinv,
                                 const float* __restrict__ W,
                                 float* __restrict__ H) {
    constexpr int LDSS = KDIM + 4;                 // padded stride (banks: 4m distinct mod 64)
    __shared__ __align__(16) float As[16 * LDSS];

    const int row0 = blockIdx.x * 16;
    const int tid  = threadIdx.x;

    // Cooperative load of the 16x128 A tile, scaled by dsrc_inv.
    for (int i = tid; i < 16 * KDIM; i += blockDim.x) {
        int r = i >> 7;            // /128
        int c = i & (KDIM - 1);
        As[r * LDSS + c] = X[(size_t)(row0 + r) * KDIM + c] * dinv[row0 + r];
    }
    __syncthreads();

    const int lane  = tid & 31;
    const int wave  = tid >> 5;
    const int n     = lane & 15;   // column within 16x16 tile
    const int khalf = lane >> 4;   // 0 or 1
    const int m16   = lane & 15;   // A row for this lane
    const int ncol  = wave * 16 + n;

    v8f acc = {};
#pragma unroll
    for (int k0 = 0; k0 < KDIM; k0 += 4) {
        const int ka = k0 + 2 * khalf;             // even -> 8B aligned LDS load
        v2f a = *(const v2f*)&As[m16 * LDSS + ka]; // ds_load_b64
        v2f b;
        b.x = W[(size_t)ka * FOUT + ncol];
        b.y = W[(size_t)(ka + 1) * FOUT + ncol];
        acc = __builtin_amdgcn_wmma_f32_16x16x4_f32(
            /*neg_a=*/false, a, /*neg_b=*/false, b,
            /*c_mod=*/(short)0, acc, /*reuse_a=*/false, /*reuse_b=*/false);
    }

#pragma unroll
    for (int r = 0; r < 8; ++r) {
        const int m = r + 8 * khalf;
        H[(size_t)(row0 + m) * FOUT + ncol] = acc[r];
    }
}

// ---------------------------------------------------------------------------
// AGG[dst[e]] += H[src[e]]   (one wave per edge, F/32 floats per lane)
// ---------------------------------------------------------------------------
template <int F>
__global__ void scatter_add_kernel(const float* __restrict__ H,
                                   const int* __restrict__ src,
                                   const int* __restrict__ dst,
                                   float* __restrict__ AGG, int ne) {
    const int gid  = blockIdx.x * blockDim.x + threadIdx.x;
    const int e    = gid >> 5;
    const int lane = gid & 31;
    if (e >= ne) return;

    const int s = src[e];
    const int d = dst[e];
    constexpr int V = F / 32;      // 4 (F=128) or 2 (F=64)

    const float* hp = H   + (size_t)s * F + lane * V;
    float*       op = AGG + (size_t)d * F + lane * V;

    if (V == 4) {
        float4 v = *(const float4*)hp;
        atomicAdd(op + 0, v.x);
        atomicAdd(op + 1, v.y);
        atomicAdd(op + 2, v.z);
        atomicAdd(op + 3, v.w);
    } else {
        float2 v = *(const float2*)hp;
        atomicAdd(op + 0, v.x);
        atomicAdd(op + 1, v.y);
    }
}

// ---------------------------------------------------------------------------
// Orchestration
// ---------------------------------------------------------------------------
static inline int cdiv(long a, long b) { return (int)((a + b - 1) / b); }

extern "C" void kernel_launch(void* const* d_in, const int* in_sizes, int n_in,
                              void* d_out, int out_size, void* d_ws, size_t ws_size,
                              hipStream_t stream) {
    const float* features = (const float*)d_in[0];
    const int*   src      = (const int*)d_in[1];
    const int*   dst      = (const int*)d_in[2];
    const float* W1       = (const float*)d_in[3];
    const float* b1       = (const float*)d_in[4];
    const float* W2       = (const float*)d_in[5];
    const float* b2       = (const float*)d_in[6];
    const float* W3       = (const float*)d_in[7];
    const float* b3       = (const float*)d_in[8];
    float*       out      = (float*)d_out;

    const int N = in_sizes[0] / IN_F;   // 100000 (divisible by 16)
    const int E = in_sizes[1];          // 1600000

    float* ws   = (float*)d_ws;
    float* dsrc = ws;                               // [N]
    float* ddst = ws + N;                           // [N]
    float* buf0 = ws + 2 * (size_t)N;               // [N, 128] (h)
    float* buf1 = buf0 + (size_t)N * HID_F;         // [N, 128] (agg / x)

    const long NH = (long)N * HID_F;
    const long NO = (long)N * OUT_F;
    const int  nblk = N / 16;

    // --- degree^{-1/2} for src and dst ---
    zero_kernel<<<cdiv(2L * N, 256), 256, 0, stream>>>(dsrc, 2L * N);
    degree_kernel<<<cdiv(E, 256), 256, 0, stream>>>(src, dst, dsrc, ddst, E);
    rsqrt_kernel<<<cdiv(2L * N, 256), 256, 0, stream>>>(dsrc, 2 * N);

    // --- layer 1: x -> buf1 ---
    gemm_rows16_wmma<HID_F><<<nblk, HID_F * 2, 0, stream>>>(features, dsrc, W1, buf0);
    zero_kernel<<<cdiv(NH, 256), 256, 0, stream>>>(buf1, NH);
    scatter_add_kernel<HID_F><<<cdiv((long)E * 32, 256), 256, 0, stream>>>(buf0, src, dst, buf1, E);
    epilogue_kernel<<<cdiv(NH, 256), 256, 0, stream>>>(buf1, ddst, b1, HID_F, NH, 1);

    // --- layer 2: buf1 -> buf1 ---
    gemm_rows16_wmma<HID_F><<<nblk, HID_F * 2, 0, stream>>>(buf1, dsrc, W2, buf0);
    zero_kernel<<<cdiv(NH, 256), 256, 0, stream>>>(buf1, NH);
    scatter_add_kernel<HID_F><<<cdiv((long)E * 32, 256), 256, 0, stream>>>(buf0, src, dst, buf1, E);
    epilogue_kernel<<<cdiv(NH, 256), 256, 0, stream>>>(buf1, ddst, b2, HID_F, NH, 1);

    // --- layer 3: buf1 -> out ---
    gemm_rows16_wmma<OUT_F><<<nblk, OUT_F * 2, 0, stream>>>(buf1, dsrc, W3, buf0);
    zero_kernel<<<cdiv(NO, 256), 256, 0, stream>>>(out, NO);
    scatter_add_kernel<OUT_F><<<cdiv((long)E * 32, 256), 256, 0, stream>>>(buf0, src, dst, out, E);
    epilogue_kernel<<<cdiv(NO, 256), 256, 0, stream>>>(out, ddst, b3, OUT_F, NO, 0);
}